// FusedKQnA_1726576854813
// MI455X (gfx1250) — compile-verified
//
#include <hip/hip_runtime.h>
#include <hip/hip_bf16.h>
#include <math.h>

// ---- problem constants (from reference) ----
#define NQ      4
#define KS      3
#define STRIDEV 2
#define PADV    1
#define Bsz     16
#define Cin     128
#define Himg    56
#define Wimg    56
#define HC      32            // head channels
#define HP      8             // N_HEADS * STRIDE
#define CS      256           // Cin * STRIDE
#define Npix    (Himg * Wimg) // 3136
#define ROWS    (Bsz * Npix)  // 50176
#define HO      28
#define WO      28
#define POUT    (HO * WO)     // 784

typedef __attribute__((ext_vector_type(2))) float v2f;
typedef __attribute__((ext_vector_type(8))) float v8f;

__device__ __forceinline__ v8f wmma4(v2f a, v2f b, v8f c) {
    return __builtin_amdgcn_wmma_f32_16x16x4_f32(
        false, a, false, b, (short)0, c, false, false);
}

// =====================================================================
// Kernel 1: K = xf@Wk, V = xf@Wv   (f32 WMMA 16x16x4, exact fp32 math)
// One wave per 16x64 output tile: the (expensive, Npix-strided) A fragment
// is loaded once and reused by 4 WMMAs -> 10 loads / 4 wmma per K-step,
// and 4 independent accumulator chains pipeline on the XDL.
// A-frag (16x4 f32): lane m = lane&15; lanes<16 hold K={0,1}, lanes>=16 K={2,3}.
// C/D  (16x16 f32): VGPR j = row j (lanes 0-15) / row 8+j (lanes 16-31).
// =====================================================================
__global__ __launch_bounds__(256) void kv_gemm_wmma(
    const float* __restrict__ x,  const float* __restrict__ Wk,
    const float* __restrict__ Wv, float* __restrict__ Kbuf,
    float* __restrict__ Vbuf)
{
    const int lane  = threadIdx.x & 31;
    const int wid   = blockIdx.x * 8 + (threadIdx.x >> 5);
    const int cg    = wid & 3;                   // col-group of 64
    const int which = (wid >> 2) & 1;            // 0 = K, 1 = V
    const int rtile = wid >> 3;                  // 0..3135

    const float* W   = which ? Wv   : Wk;
    float*       Out = which ? Vbuf : Kbuf;

    const int half = lane >> 4;                  // 0 or 1
    const int l15  = lane & 15;
    const int row  = rtile * 16 + l15;           // global A row (= M index)
    const int b    = row / Npix;                 // tiles never straddle b (3136%16==0)
    const int n    = row - b * Npix;
    const int col0 = cg * 64 + l15;              // first B/D column for this lane

    const float* Ap = x + (size_t)b * Cin * Npix + n;   // + c*Npix walks channels
    const float* Wp = W + col0;

    v8f acc0 = {}, acc1 = {}, acc2 = {}, acc3 = {};
    for (int kk = 0; kk < Cin; kk += 4) {
        const int kc = kk + half * 2;
        v2f a;
        a.x = Ap[(size_t)(kc)     * Npix];
        a.y = Ap[(size_t)(kc + 1) * Npix];
        const float* w0 = Wp + (size_t)kc * CS;
        v2f b0, b1, b2, b3;
        b0.x = w0[0];   b0.y = w0[CS];
        b1.x = w0[16];  b1.y = w0[CS + 16];
        b2.x = w0[32];  b2.y = w0[CS + 32];
        b3.x = w0[48];  b3.y = w0[CS + 48];
        if (kk + 4 < Cin) {
            __builtin_prefetch(&Ap[(size_t)(kc + 4) * Npix], 0, 0);
            __builtin_prefetch(&w0[4 * CS], 0, 0);
        }
        acc0 = wmma4(a, b0, acc0);
        acc1 = wmma4(a, b1, acc1);
        acc2 = wmma4(a, b2, acc2);
        acc3 = wmma4(a, b3, acc3);
    }

    const int orow = rtile * 16 + half * 8;
    #pragma unroll
    for (int j = 0; j < 8; ++j) {
        float* o = Out + (size_t)(orow + j) * CS + col0;
        o[0]  = acc0[j];
        o[16] = acc1[j];
        o[32] = acc2[j];
        o[48] = acc3[j];
    }
}

// =====================================================================
// Kernel 2: cost[b,n,q*8+h] = exp( (q_param[q,h,:]·k[b,n,h,:]) / sqrt(hc) )
// The global max-subtraction in the reference cancels in num/den -> skipped.
// =====================================================================
__global__ __launch_bounds__(256) void cost_kernel(
    const float* __restrict__ Kbuf, const float* __restrict__ qparam,
    float* __restrict__ cost)
{
    const int idx = blockIdx.x * 256 + threadIdx.x;   // row*32 + qh
    const int qh  = idx & 31;
    const int row = idx >> 5;
    const int q   = qh >> 3, h = qh & 7;

    const float* kp = Kbuf   + (size_t)row * CS + h * HC;
    const float* qp = qparam + (size_t)q   * CS + h * HC;
    float s = 0.f;
    #pragma unroll
    for (int c = 0; c < HC; ++c) s = fmaf(qp[c], kp[c], s);
    cost[idx] = __expf(s * 0.17677669529663687f);     // 1/sqrt(32)
}

// =====================================================================
// Kernel 3: fused 3x3 stride-2 depthwise num/den convolution + sum_q num/den.
// One block per output pixel (b,ho,wo); thread t -> channel (h = t>>5, c = t&31).
// Kernel weights and the 9x32 cost taps staged in LDS once.
// pre layout: (B, Cs, Ho*Wo) so the projection GEMM reads contiguous columns.
// =====================================================================
__global__ __launch_bounds__(256) void na_conv_kernel(
    const float* __restrict__ Vbuf, const float* __restrict__ cost,
    const float* __restrict__ attn_scale, const float* __restrict__ rpb,
    float* __restrict__ pre)
{
    __shared__ float knum [KS * KS * 32];
    __shared__ float kden [KS * KS * 32];
    __shared__ float scost[KS * KS * 32];

    const int t  = threadIdx.x;
    const int bi = blockIdx.x;
    const int b  = bi / POUT;
    const int p  = bi - b * POUT;
    const int ho = p / WO;
    const int wo = p - ho * WO;

    for (int i = t; i < KS * KS * 32; i += 256) {
        const float re = __expf(rpb[i]);     // exp(rpb - max) scale cancels in ratio
        kden[i] = re;
        knum[i] = re * attn_scale[i];
        // stage cost for all 9 taps (0 for out-of-range -> contributes nothing)
        const int kk = i >> 5, qh = i & 31;
        const int ki = kk / KS, kj = kk - ki * KS;
        const int hi = ho * STRIDEV - PADV + ki;
        const int wi = wo * STRIDEV - PADV + kj;
        float cv = 0.f;
        if (hi >= 0 && hi < Himg && wi >= 0 && wi < Wimg)
            cv = cost[(((size_t)(b * Npix + hi * Wimg + wi)) << 5) + qh];
        scost[i] = cv;
    }
    __syncthreads();

    const int h = t >> 5;
    const int c = t & 31;
    float num[NQ] = {0.f, 0.f, 0.f, 0.f};
    float den[NQ] = {0.f, 0.f, 0.f, 0.f};

    #pragma unroll
    for (int ki = 0; ki < KS; ++ki) {
        const int hi = ho * STRIDEV - PADV + ki;
        #pragma unroll
        for (int kj = 0; kj < KS; ++kj) {
            const int wi = wo * STRIDEV - PADV + kj;
            const int kk = ki * KS + kj;
            float vv = 0.f;
            if (hi >= 0 && hi < Himg && wi >= 0 && wi < Wimg)
                vv = Vbuf[(size_t)(b * Npix + hi * Wimg + wi) * CS + h * HC + c];
            #pragma unroll
            for (int q = 0; q < NQ; ++q) {
                const int qh = (q << 3) | h;
                const float cv = scost[(kk << 5) + qh];
                num[q] = fmaf(knum[(kk << 5) + qh] * cv, vv, num[q]);
                den[q] = fmaf(kden[(kk << 5) + qh],      cv, den[q]);
            }
        }
    }
    float o = 0.f;
    #pragma unroll
    for (int q = 0; q < NQ; ++q) o += num[q] / den[q];   // den>0 always (exp>0)
    pre[(size_t)(b * CS + h * HC + c) * POUT + p] = o;
}

// =====================================================================
// Kernel 4: out[b,o,p] = sum_c Wout[o,c] * pre[b,c,p]  (f32 WMMA 16x16x4)
// One wave per 64x16 tile: the (POUT-strided) pre B-fragment is loaded once
// and reused by 4 WMMAs across 4 o-tiles.
// =====================================================================
__global__ __launch_bounds__(256) void proj_gemm_wmma(
    const float* __restrict__ Wout, const float* __restrict__ pre,
    float* __restrict__ out)
{
    const int lane  = threadIdx.x & 31;
    const int wid   = blockIdx.x * 8 + (threadIdx.x >> 5);
    const int og    = wid & 3;                  // o-group of 64 rows
    const int ptile = wid >> 2;                 // 0..783
    const int b     = ptile / (POUT / 16);      // 49 pixel tiles per image
    const int pt    = ptile - b * (POUT / 16);
    const int pbase = pt * 16;

    const int half  = lane >> 4;
    const int l15   = lane & 15;
    const int orow0 = og * 64 + l15;            // first A row for this lane

    const float* Bp = pre + (size_t)b * CS * POUT + pbase + l15;

    v8f acc0 = {}, acc1 = {}, acc2 = {}, acc3 = {};
    for (int kk = 0; kk < CS; kk += 4) {
        const int kc = kk + half * 2;
        v2f bf;
        bf.x = Bp[(size_t)(kc)     * POUT];
        bf.y = Bp[(size_t)(kc + 1) * POUT];
        const float* w0 = Wout + (size_t)orow0 * CS + kc;
        v2f a0, a1, a2, a3;
        a0.x = w0[0];       a0.y = w0[1];
        a1.x = w0[16 * CS]; a1.y = w0[16 * CS + 1];
        a2.x = w0[32 * CS]; a2.y = w0[32 * CS + 1];
        a3.x = w0[48 * CS]; a3.y = w0[48 * CS + 1];
        if (kk + 4 < CS)
            __builtin_prefetch(&Bp[(size_t)(kc + 4) * POUT], 0, 0);
        acc0 = wmma4(a0, bf, acc0);
        acc1 = wmma4(a1, bf, acc1);
        acc2 = wmma4(a2, bf, acc2);
        acc3 = wmma4(a3, bf, acc3);
    }

    const int ob = og * 64 + half * 8;
    #pragma unroll
    for (int j = 0; j < 8; ++j) {
        float* o = out + (size_t)(b * CS + ob + j) * POUT + pbase + l15;
        o[0]         = acc0[j];
        o[16 * POUT] = acc1[j];
        o[32 * POUT] = acc2[j];
        o[48 * POUT] = acc3[j];
    }
}

// =====================================================================
extern "C" void kernel_launch(void* const* d_in, const int* in_sizes, int n_in,
                              void* d_out, int out_size, void* d_ws, size_t ws_size,
                              hipStream_t stream) {
    const float* x          = (const float*)d_in[0];
    const float* Wk         = (const float*)d_in[1];
    const float* Wv         = (const float*)d_in[2];
    const float* Wout       = (const float*)d_in[3];
    const float* qparam     = (const float*)d_in[4];
    const float* attn_scale = (const float*)d_in[5];
    const float* rpb        = (const float*)d_in[6];

    float* ws   = (float*)d_ws;
    float* Kbuf = ws;                                    // ROWS*CS     floats
    float* Vbuf = Kbuf + (size_t)ROWS * CS;              // ROWS*CS     floats
    float* cst  = Vbuf + (size_t)ROWS * CS;              // ROWS*32     floats
    float* pre  = cst  + (size_t)ROWS * 32;              // Bsz*CS*POUT floats
    float* out  = (float*)d_out;

    // K1: 3136 rtiles * 4 col-groups * 2 (K,V) = 25088 waves -> 3136 blocks
    kv_gemm_wmma<<<3136 * 4 * 2 / 8, 256, 0, stream>>>(x, Wk, Wv, Kbuf, Vbuf);
    // K2: ROWS*32 threads
    cost_kernel<<<((size_t)ROWS * 32) / 256, 256, 0, stream>>>(Kbuf, qparam, cst);
    // K3: one block per output pixel
    na_conv_kernel<<<Bsz * POUT, 256, 0, stream>>>(Vbuf, cst, attn_scale, rpb, pre);
    // K4: 4 o-groups * 784 pixel tiles = 3136 waves -> 392 blocks
    proj_gemm_wmma<<<4 * 784 / 8, 256, 0, stream>>>(Wout, pre, out);
}